// NeuralSparse_Sampler_20916490731927
// MI455X (gfx1250) — compile-verified
//
#include <hip/hip_runtime.h>

// NeuralSparse Gumbel edge sampler for MI455X (gfx1250, wave32).
//
// Per edge: concat(h[row],h[col]) (256 f32) @ W1 (256x64) -> ReLU -> @ W2 -> logit
//           weight = (logit + g1 > g0), g = -log(-log(u)).
//
// Strategy: fp32 WMMA (V_WMMA_F32_16X16X4_F32). One wave computes 16 edges
// (a 16x64 hidden tile = 4 accumulators, K=256 in steps of 4). A-fragments are
// 8-byte global loads from L2-resident h; B-fragments come from a per-block
// pre-swizzled copy of W1^T in LDS (64KB), read conflict-free with immediate
// DS offsets. Epilogue does ReLU + W2 dot + shfl_xor column reduction + Gumbel.

typedef __attribute__((ext_vector_type(2))) float v2f;
typedef __attribute__((ext_vector_type(8))) float v8f;

#define Hdim 128
#define HID  64
#define KBn  64                 // 256 / 4 k-steps
#define WAVES_PER_BLOCK 8
#define EDGES_PER_BLOCK (WAVES_PER_BLOCK * 16)

__global__ __launch_bounds__(256) void ns_sampler_kernel(
    const float* __restrict__ h,
    const float* __restrict__ W1,
    const float* __restrict__ b1,
    const float* __restrict__ W2,
    const float* __restrict__ b2,
    const float* __restrict__ u,
    const int*   __restrict__ ei,
    float* __restrict__ out,
    int E)
{
    // Pre-swizzled B fragments: Bfrag[(kb*4 + nt)*32 + lane] = (W1[k0][n], W1[k0+1][n])
    // with n = nt*16 + lane%16, k0 = kb*4 + 2*(lane/16).  64 KB.
    __shared__ v2f Bfrag[KBn * 4 * 32];

    const int tid = threadIdx.x;

    for (int i = tid; i < KBn * 4 * 32; i += 256) {
        int l  = i & 31;
        int nt = (i >> 5) & 3;
        int kb = i >> 7;
        int n  = nt * 16 + (l & 15);
        int k0 = kb * 4 + ((l >> 4) << 1);
        v2f v;
        v.x = W1[k0 * HID + n];
        v.y = W1[(k0 + 1) * HID + n];
        Bfrag[i] = v;
    }
    __syncthreads();

    const int lane  = tid & 31;
    const int wave  = tid >> 5;
    const int ln    = lane & 15;      // column / M-row selector
    const int khalf = lane >> 4;      // which K pair within a k-step

    const int edgeBase = (blockIdx.x * WAVES_PER_BLOCK + wave) * 16;

    // This lane supplies A rows for edge M = ln of the tile.
    int eM  = edgeBase + ln;
    int eMc = eM < E ? eM : (E - 1);          // clamp: keep EXEC all-ones for WMMA
    int rowNode = ei[eMc];
    int colNode = ei[E + eMc];
    const float* hrow = h + (long)rowNode * Hdim + khalf * 2;
    const float* hcol = h + (long)colNode * Hdim + khalf * 2;

    // per-lane bias / W2 slices (n = nt*16 + ln)
    float b1v[4], w2v[4];
#pragma unroll
    for (int nt = 0; nt < 4; ++nt) {
        b1v[nt] = b1[nt * 16 + ln];
        w2v[nt] = W2[nt * 16 + ln];
    }
    const float b2v = b2[0];

    v8f acc[4] = {};                          // 4 N-tiles of the 16x64 hidden block
    const v2f* bf = &Bfrag[lane];

    // K = 0..127  -> row-node features
#pragma unroll 4
    for (int kb = 0; kb < 32; ++kb) {
        v2f a = *(const v2f*)(hrow + kb * 4);
#pragma unroll
        for (int nt = 0; nt < 4; ++nt) {
            acc[nt] = __builtin_amdgcn_wmma_f32_16x16x4_f32(
                false, a, false, bf[(kb * 4 + nt) * 32],
                (short)0, acc[nt], false, false);
        }
    }
    // K = 128..255 -> col-node features
#pragma unroll 4
    for (int kb = 32; kb < 64; ++kb) {
        v2f a = *(const v2f*)(hcol + (kb - 32) * 4);
#pragma unroll
        for (int nt = 0; nt < 4; ++nt) {
            acc[nt] = __builtin_amdgcn_wmma_f32_16x16x4_f32(
                false, a, false, bf[(kb * 4 + nt) * 32],
                (short)0, acc[nt], false, false);
        }
    }

    // Epilogue: hid = relu(acc + b1); partial = hid . W2 ; reduce over 16 columns.
    // C layout: acc[nt][j] holds row M = j + 8*khalf, col N = nt*16 + ln.
    float red[8];
#pragma unroll
    for (int j = 0; j < 8; ++j) {
        float s = 0.f;
#pragma unroll
        for (int nt = 0; nt < 4; ++nt)
            s += fmaxf(acc[nt][j] + b1v[nt], 0.f) * w2v[nt];
        s += __shfl_xor(s, 1, 32);
        s += __shfl_xor(s, 2, 32);
        s += __shfl_xor(s, 4, 32);
        s += __shfl_xor(s, 8, 32);
        red[j] = s;                            // logit (pre-b2) of row j + 8*khalf
    }

    if (ln < 8) {
        float myv = red[0];
#pragma unroll
        for (int j = 1; j < 8; ++j)
            if (ln == j) myv = red[j];
        int r = ln + 8 * khalf;
        int e = edgeBase + r;
        if (e < E) {
            float logit = myv + b2v;
            float u0 = u[2 * e];
            float u1 = u[2 * e + 1];
            float g0 = -logf(-logf(u0));
            float g1 = -logf(-logf(u1));
            // argmax([g0, logit+g1]) == 1  (ties -> 0, matching jnp.argmax)
            out[e]     = (logit + g1) > g0 ? 1.0f : 0.0f;
            out[E + e] = logit;
        }
    }
}

extern "C" void kernel_launch(void* const* d_in, const int* in_sizes, int n_in,
                              void* d_out, int out_size, void* d_ws, size_t ws_size,
                              hipStream_t stream) {
    const float* h  = (const float*)d_in[0];
    const float* W1 = (const float*)d_in[1];
    const float* b1 = (const float*)d_in[2];
    const float* W2 = (const float*)d_in[3];
    const float* b2 = (const float*)d_in[4];
    const float* u  = (const float*)d_in[5];
    const int*   ei = (const int*)d_in[6];

    const int E = in_sizes[6] / 2;            // edge_index is (2, E)
    const int blocks = (E + EDGES_PER_BLOCK - 1) / EDGES_PER_BLOCK;

    ns_sampler_kernel<<<blocks, 256, 0, stream>>>(
        h, W1, b1, W2, b2, u, ei, (float*)d_out, E);
}